// lveg_9698036154934
// MI455X (gfx1250) — compile-verified
//
#include <hip/hip_runtime.h>
#include <hip/hip_bf16.h>
#include <math.h>

typedef float v2f __attribute__((ext_vector_type(2)));
typedef float v4f __attribute__((ext_vector_type(4)));
typedef float v8f __attribute__((ext_vector_type(8)));

#define LOG2PI 1.837877f
#define NL 46
#define KP 4
#define NR (NL * KP)        // 184 mixture rows (r = c*4 + k)
#define PPAD 48             // parent dim padded to 3 WMMA col-tiles
#define PL (NL * PPAD)      // plane stride in floats (2208)
#define NEGBIG -1e30f

// packed workspace layout (float offsets); total 13*PL floats = 114,816 B
#define WS_BN  0            // float4/cp: (P0, P1, l11, P3)  -> num-GEMM B
#define WS_BD  (4 * PL)     // float4/cp: (detL1, 0, 0, l11) -> det-GEMM B
#define WS_C0  (8 * PL)     // zeta1 + t_weight + LOG2PI
#define WS_REC (9 * PL)     // float4/cp: (l00, l01, e1_0, e1_1)

// ---------------------------------------------------------------------------
// Precompute all (child c, parent p) position-invariant Gaussian algebra.
// ---------------------------------------------------------------------------
__global__ void lveg_precomp(const float* __restrict__ tW,
                             const float* __restrict__ tMu,
                             const float* __restrict__ tVarP,
                             float* __restrict__ ws) {
    int i = blockIdx.x * blockDim.x + threadIdx.x;   // i = c*PPAD + p
    if (i >= NL * PPAD) return;
    int c = i / PPAD, p = i % PPAD;
    float P0 = 0.f, P1 = 0.f, P3 = 0.f, l11 = 0.f, INV = 0.f, C0 = 0.f;
    float l00 = 0.f, l01 = 0.f, e0 = 0.f, e1 = 0.f;
    if (p < NL) {
        int cp = c * NL + p;
        float v0 = tVarP[cp * 3 + 0], v1 = tVarP[cp * 3 + 1], v2 = tVarP[cp * 3 + 2];
        float S00 = v0 * v0 + v1 * v1;
        float S01 = v1 * v2;
        float S11 = v2 * v2;
        float detS = S00 * S11 - S01 * S01;
        float invd = 1.0f / detS;
        l00 = S11 * invd; l01 = -S01 * invd; l11 = S00 * invd;
        float tm0 = tMu[cp * 2 + 0], tm1 = tMu[cp * 2 + 1];
        e0 = l00 * tm0 + l01 * tm1;
        e1 = l01 * tm0 + l11 * tm1;
        float q1 = e0 * (S00 * e0 + S01 * e1) + e1 * (S01 * e0 + S11 * e1);
        float zeta1 = -0.5f * (2.0f * LOG2PI + logf(detS) + q1);
        P0  = e0 * e0 * l11 - 2.0f * l01 * e0 * e1 + e1 * e1 * l00;
        P1  = 2.0f * (e0 * l11 - l01 * e1);
        P3  = e1 * e1;
        INV = l00 * l11 - l01 * l01;
        C0  = zeta1 + tW[cp] + LOG2PI;
    }
    ws[WS_BN + 4 * i + 0] = P0;  ws[WS_BN + 4 * i + 1] = P1;
    ws[WS_BN + 4 * i + 2] = l11; ws[WS_BN + 4 * i + 3] = P3;
    ws[WS_BD + 4 * i + 0] = INV; ws[WS_BD + 4 * i + 1] = 0.f;
    ws[WS_BD + 4 * i + 2] = 0.f; ws[WS_BD + 4 * i + 3] = l11;
    ws[WS_C0 + i] = C0;
    ws[WS_REC + 4 * i + 0] = l00; ws[WS_REC + 4 * i + 1] = l01;
    ws[WS_REC + 4 * i + 2] = e0;  ws[WS_REC + 4 * i + 3] = e1;
}

// ---------------------------------------------------------------------------
// Main scan: one block per batch element, whole recurrence resident in LDS.
// Phase 2: V_WMMA_F32_16X16X4_F32 — per child c, num/det over (k,p) are K=4
// GEMMs with shared A rows (1, a, a^2, lam2). B fragments are single b64
// loads from the packed planes (address-selected per lane half, no branches).
// ---------------------------------------------------------------------------
__global__ __launch_bounds__(256) void lveg_main(
        const int*   __restrict__ tokens,
        const float* __restrict__ sWt,
        const float* __restrict__ sMt,
        const float* __restrict__ sVt,
        const float* __restrict__ ws,
        float* __restrict__ out, int L) {
    __shared__ float sScale[NL * NR];               // [p][r] pruning candidates
    __shared__ float sSC[NR], sMUc[NR], sVAc[NR];   // carry, r = c*4 + k
    __shared__ float sAf[NR], sL2[NR], sZ[NR];      // step features
    __shared__ float sew[NL], sem[NL], sev[NL];     // emission at position t
    __shared__ float sCandV[NL * 16];               // stage-1 top-k candidates
    __shared__ int   sCandI[NL * 16];

    const int b = blockIdx.x, tid = threadIdx.x;
    const int lane = tid & 31, wave = tid >> 5;
    const int m = lane & 15;
    const bool hi = lane >= 16;

    // ---- init carry with emission at position 0 ----
    if (tid < NL) {
        int p = tid;
        int tok = tokens[b * L];
        sSC[p * 4]  = sWt[tok * NL + p];
        sMUc[p * 4] = sMt[tok * NL + p];
        float v = sVt[tok * NL + p];
        sVAc[p * 4] = v * v;
        for (int s = 1; s < KP; ++s) {
            sSC[p * 4 + s] = NEGBIG; sMUc[p * 4 + s] = 0.f; sVAc[p * 4 + s] = 1.f;
        }
    }
    __syncthreads();

    for (int t = 1; t < L; ++t) {
        // ---- phase 1: per-row features + emission gather (+ prefetch next) ----
        if (tid < NR) {
            float gm = sMUc[tid], gv = sVAc[tid];
            float l2 = 1.0f / gv;
            float a  = gm * l2;
            sAf[tid] = a;
            sL2[tid] = l2;
            sZ[tid]  = sSC[tid] - 0.5f * (LOG2PI + logf(gv) + gm * a);
        } else if (tid < NR + NL) {
            int p = tid - NR;
            int tok = tokens[b * L + t];
            sew[p] = sWt[tok * NL + p];
            sem[p] = sMt[tok * NL + p];
            float v = sVt[tok * NL + p];
            sev[p] = v * v;
            if (p == 0 && t + 1 < L) {   // hide next step's gather latency
                int tn = tokens[b * L + t + 1];
                __builtin_prefetch(&sWt[tn * NL], 0, 0);
                __builtin_prefetch(&sMt[tn * NL], 0, 0);
                __builtin_prefetch(&sVt[tn * NL], 0, 0);
            }
        }
        __syncthreads();

        // ---- phase 2: WMMA per (child c, parent tile) ----
        // A (16x4 f32): lane m=lane%16; VGPR0 = K{0|2}, VGPR1 = K{1|3}
        // B (4x16 f32): lane n=lane%16; VGPR0 = K{0|2}, VGPR1 = K{1|3}
        for (int item = wave; item < NL * 3; item += 8) {
            int c = item / 3, pt = item % 3;
            v2f A;
            if (m < KP) {
                int r = c * 4 + m;
                float a = sAf[r], l2 = sL2[r];
                A.x = hi ? a * a : 1.0f;   // features (1, a, a^2, lam2)
                A.y = hi ? l2    : a;
            } else { A.x = 0.f; A.y = 0.f; }
            int p   = pt * 16 + m;          // B column for this lane (n == m)
            int idx = c * PPAD + p;         // padded planes: always in range
            int sel = hi ? 2 : 0;           // lane-half selects K{2,3} vs K{0,1}
            v2f Bn = *(const v2f*)(ws + WS_BN + 4 * idx + sel);
            v2f Bd = *(const v2f*)(ws + WS_BD + 4 * idx + sel);
            v8f z = {};
            v8f dn = __builtin_amdgcn_wmma_f32_16x16x4_f32(
                false, A, false, Bn, (short)0, z, false, false);
            v8f dd = __builtin_amdgcn_wmma_f32_16x16x4_f32(
                false, A, false, Bd, (short)0, z, false, false);
            if (!hi && p < NL) {            // rows k=0..3 -> VGPRs 0..3, lanes 0..15
                float c0 = ws[WS_C0 + idx];
#pragma unroll
                for (int j = 0; j < KP; ++j) {
                    float num = dn[j], det = dd[j];
                    int r = c * 4 + j;
                    sScale[p * NR + r] = c0 + sZ[r] + 0.5f * (num / det - logf(det));
                }
            }
        }
        __syncthreads();

        // ---- phase 3a: partial top-4 (4 threads per parent, stride-4 slices) ----
        if (tid < NR) {
            int p = tid >> 2, q = tid & 3;
            const float* row = &sScale[p * NR];
            float v0 = -3e38f, v1 = -3e38f, v2 = -3e38f, v3 = -3e38f;
            int i0 = 0, i1 = 0, i2 = 0, i3 = 0;
            for (int r = q; r < NR; r += 4) {
                float v = row[r];
                if (v > v3) {
                    if (v > v0)      { v3=v2;i3=i2; v2=v1;i2=i1; v1=v0;i1=i0; v0=v;i0=r; }
                    else if (v > v1) { v3=v2;i3=i2; v2=v1;i2=i1; v1=v;i1=r; }
                    else if (v > v2) { v3=v2;i3=i2; v2=v;i2=r; }
                    else             { v3=v;i3=r; }
                }
            }
            int base = tid * 4;
            sCandV[base + 0] = v0; sCandI[base + 0] = i0;
            sCandV[base + 1] = v1; sCandI[base + 1] = i1;
            sCandV[base + 2] = v2; sCandI[base + 2] = i2;
            sCandV[base + 3] = v3; sCandI[base + 3] = i3;
        }
        __syncthreads();

        // ---- phase 3b: merge 16 candidates, recompute survivors, emission ----
        if (tid < NL) {
            int p = tid;
            float v0 = -3e38f, v1 = -3e38f, v2 = -3e38f, v3 = -3e38f;
            int i0 = 0, i1 = 0, i2 = 0, i3 = 0;
            int base = p * 16;
            for (int u = 0; u < 16; ++u) {
                float v = sCandV[base + u];
                int   r = sCandI[base + u];
                if (v > v3) {
                    if (v > v0)      { v3=v2;i3=i2; v2=v1;i2=i1; v1=v0;i1=i0; v0=v;i0=r; }
                    else if (v > v1) { v3=v2;i3=i2; v2=v1;i2=i1; v1=v;i1=r; }
                    else if (v > v2) { v3=v2;i3=i2; v2=v;i2=r; }
                    else             { v3=v;i3=r; }
                }
            }
            float tv[KP] = {v0, v1, v2, v3};
            int   ti[KP] = {i0, i1, i2, i3};
            float emv = sem[p], evv = sev[p], ewv = sew[p];
            float inv_ev = 1.0f / evv;
            float ze = -0.5f * (LOG2PI + logf(evv) + emv * emv * inv_ev);
            for (int s = 0; s < KP; ++s) {
                int r  = ti[s];
                int cs = r >> 2;
                int idx = cs * PPAD + p;
                float l2 = sL2[r], a = sAf[r];
                v4f rc = *(const v4f*)(ws + WS_REC + 4 * idx);  // l00,l01,e10,e11
                float l11 = ws[WS_BN + 4 * idx + 2];
                float lm00 = rc.x + l2;
                float det  = lm00 * l11 - rc.y * rc.y;
                float inv  = 1.0f / det;
                float em0  = rc.z + a;
                float mu2  = (lm00 * rc.w - rc.y * em0) * inv;  // marginal over parent
                float var2 = lm00 * inv;
                // _gauss1d_multi with emission (sem, sev)
                float iv2 = 1.0f / var2;
                float lam = iv2 + inv_ev;
                float vm  = 1.0f / lam;
                float emm = mu2 * iv2 + emv * inv_ev;
                float zz  = -0.5f * (LOG2PI + logf(var2) + mu2 * mu2 * iv2) + ze
                            + 0.5f * (LOG2PI + logf(vm) + emm * emm * vm);
                sSC[p * 4 + s]  = tv[s] + zz + ewv;
                sMUc[p * 4 + s] = vm * emm;
                sVAc[p * 4 + s] = vm;
            }
        }
        __syncthreads();
    }

    // ---- final inside score: logsumexp over K*NL components ----
    if (tid == 0) {
        float mx = -3e38f;
        for (int r = 0; r < NR; ++r) mx = fmaxf(mx, sSC[r]);
        float sum = 0.f;
        for (int r = 0; r < NR; ++r) sum += expf(sSC[r] - mx);
        out[b] = logf(sum) + mx;
    }
}

// ---------------------------------------------------------------------------
extern "C" void kernel_launch(void* const* d_in, const int* in_sizes, int n_in,
                              void* d_out, int out_size, void* d_ws, size_t ws_size,
                              hipStream_t stream) {
    (void)n_in; (void)out_size; (void)ws_size;
    const int*   tokens = (const int*)d_in[0];
    const float* sWt    = (const float*)d_in[1];
    const float* sMt    = (const float*)d_in[2];
    const float* sVt    = (const float*)d_in[3];
    const float* tW     = (const float*)d_in[4];
    const float* tMu    = (const float*)d_in[5];
    const float* tVarP  = (const float*)d_in[6];
    float* ws  = (float*)d_ws;     // needs 13*PL*4 = 114,816 bytes
    float* out = (float*)d_out;
    const int B = 64;
    const int L = in_sizes[0] / B;

    lveg_precomp<<<(NL * PPAD + 255) / 256, 256, 0, stream>>>(tW, tMu, tVarP, ws);
    lveg_main<<<B, 256, 0, stream>>>(tokens, sWt, sMt, sVt, ws, out, L);
}